// PositionalEmbedding_75685913690611
// MI455X (gfx1250) — compile-verified
//
#include <hip/hip_runtime.h>

typedef __attribute__((ext_vector_type(2))) float v2f;
typedef __attribute__((ext_vector_type(8))) float v8f;

#define EMB 1024

// sin/cos range-reduction constants
#define INV_2PI   0.15915493667125702f   // 1/(2*pi), round-to-nearest float
#define PI2_HI    6.28125f               // 8-bit mantissa: k*PI2_HI exact for k < 2^15
#define PI2_MID   1.9353071795e-3f       // 2*pi - PI2_HI

__global__ __launch_bounds__(256)
void posemb_wmma_kernel(const float* __restrict__ div, float* __restrict__ out) {
    const int lane = threadIdx.x & 31;
    const int wave = threadIdx.x >> 5;
    const int bid  = blockIdx.x;

    const int tile_m = bid >> 3;                 // 16-row tile index      [0, 2048)
    const int tile_n = ((bid & 7) << 3) + wave;  // 16-col tile index      [0, 64)
    const int row0 = tile_m << 4;
    const int col0 = tile_n << 4;

    // Rank-1 outer product pos[16] (x) div[16] via V_WMMA_F32_16X16X4_F32
    // using only the K=0 slice of A (16x4) and B (4x16).
    // A layout (32-bit, 16x4): lanes 0-15 -> {K0,K1}, lanes 16-31 -> {K2,K3}, M = lane%16.
    v2f a;
    a.x = (lane < 16) ? (float)(row0 + lane) : 0.0f;
    a.y = 0.0f;
    // B layout (32-bit, 4x16): lanes 0-15 -> rows K0 (v0), K1 (v1); lanes 16-31 -> K2,K3; N = lane%16.
    v2f b;
    b.x = (lane < 16) ? div[col0 + lane] : 0.0f;
    b.y = 0.0f;
    v8f c = {};
    // D[m][n] = pos[m] * div[n]; C/D layout: VGPR v, lane l -> M = v + 8*(l>=16), N = l&15
    v8f pe = __builtin_amdgcn_wmma_f32_16x16x4_f32(
        /*neg_a=*/false, a, /*neg_b=*/false, b,
        /*c_mod=*/(short)0, c, /*reuse_a=*/false, /*reuse_b=*/false);

    const int col   = col0 + (lane & 15);
    const int rbase = row0 + ((lane >> 4) << 3);

    #pragma unroll
    for (int v = 0; v < 8; ++v) {
        float x = pe[v];                         // x in [0, 32768)
        // Faithful quirk: sin(x) iff fmod(x, 2) == 0, else cos(x).
        // Exact fmod by power-of-two: x - 2*trunc(x*0.5)
        float parity = x - 2.0f * truncf(x * 0.5f);
        // Cody-Waite reduction to f in [-pi, pi] (k < 2^13 -> k*PI2_HI exact)
        float k = rintf(x * INV_2PI);
        float f = __builtin_fmaf(-k, PI2_HI, x);
        f = __builtin_fmaf(-k, PI2_MID, f);
        // v_sin_f32 takes revolutions; cos(x) = sin(x + 0.25 rev), 0.25 exact.
        float rev = f * INV_2PI + ((parity == 0.0f) ? 0.0f : 0.25f);
        float val = __builtin_amdgcn_sinf(rev);
        // Streamed output: never re-read, bypass-friendly nontemporal store.
        __builtin_nontemporal_store(val, &out[(size_t)(rbase + v) * EMB + col]);
    }
}

extern "C" void kernel_launch(void* const* d_in, const int* in_sizes, int n_in,
                              void* d_out, int out_size, void* d_ws, size_t ws_size,
                              hipStream_t stream) {
    (void)in_sizes; (void)n_in; (void)d_ws; (void)ws_size;
    // inputs: d_in[0] = x (int32, unused except shape), d_in[1] = div (float32[1024])
    const float* div = (const float*)d_in[1];
    float* out = (float*)d_out;

    const int P = out_size / EMB;          // 32768 positions
    const int blocks = (P / 16) * 8;       // one 16x16 tile per wave, 8 waves/block
    posemb_wmma_kernel<<<blocks, 256, 0, stream>>>(div, out);
}